// PerDimLB_22548578304147
// MI455X (gfx1250) — compile-verified
//
#include <hip/hip_runtime.h>
#include <math.h>

// LB-MCMC RBM flip sampler, fused single kernel for gfx1250 (wave32).
// B=1024 samples, D=256 visible dims, H=32 hidden units.

typedef __attribute__((ext_vector_type(2))) float v2f;
typedef __attribute__((ext_vector_type(8))) float v8f;

#define BN 1024
#define DN 256
#define HN 32

__device__ __forceinline__ float softplusf(float z) {
  // stable: max(z,0) + log(1 + exp(-|z|))
  return fmaxf(z, 0.f) + __logf(1.f + __expf(-fabsf(z)));
}

__device__ __forceinline__ float wave_sum(float v) {
#pragma unroll
  for (int m = 16; m > 0; m >>= 1) v += __shfl_xor(v, m, 32);
  return v;
}

__global__ __launch_bounds__(512) void lbmcmc_kernel(
    const float* __restrict__ x, const float* __restrict__ W,
    const float* __restrict__ bvec, const float* __restrict__ cvec,
    const float* __restrict__ gum, const float* __restrict__ accu,
    float* __restrict__ out) {
  __shared__ float Wt[HN * DN];   // Wt[j*256 + k] = W[k*32 + j]  (32 KB)
  __shared__ float bl[DN];        // bias b
  __shared__ float hb[16][HN];    // h = xW + c per sample
  __shared__ float h2[16][HN];    // h after the proposed flip

  const int tid  = threadIdx.x;
  const int lane = tid & 31;
  const int s    = tid >> 5;              // wave id == sample-in-block
  const int m0   = blockIdx.x * 16;       // first sample of this block
  const int bidx = m0 + s;                // this wave's sample

  // ---- stage W (transposed) and b into LDS ----
  for (int idx = tid; idx < DN * HN; idx += 512) {
    int k = idx >> 5, j = idx & 31;       // W row-major [k][j]
    Wt[j * DN + k] = W[idx];
  }
  if (tid < DN) bl[tid] = bvec[tid];

  // ---- waves 0/1: h = x@W + c via V_WMMA_F32_16X16X4_F32 (M=16,N=16,K=4) ----
  if (s < 2) {
    const int n0   = s * 16;
    const int half = lane >> 4;           // K-pair selector per ISA layout
    const int lr   = lane & 15;           // row (A) / col (B,C,D)
    v8f acc = {};
#pragma unroll 4
    for (int kk = 0; kk < DN / 4; ++kk) {
      const int kA = kk * 4 + 2 * half;
      v2f av, bv;
      av.x = x[(m0 + lr) * DN + kA];      // A: 16x4 f32, row striped in-lane
      av.y = x[(m0 + lr) * DN + kA + 1];
      bv.x = W[kA * HN + n0 + lr];        // B: 4x16 f32, row striped across lanes
      bv.y = W[(kA + 1) * HN + n0 + lr];
      acc = __builtin_amdgcn_wmma_f32_16x16x4_f32(
          false, av, false, bv, (short)0, acc, false, false);
    }
    const float cn = cvec[n0 + lr];
#pragma unroll
    for (int r = 0; r < 8; ++r)           // C/D: VGPR r -> row r + 8*half
      hb[r + 8 * half][n0 + lr] = acc[r] + cn;
  }
  __syncthreads();

  // ---- per-wave MCMC step; lane covers flip dims k = lane + 32*i ----
  float xr[8], gu[8];
#pragma unroll
  for (int i = 0; i < 8; ++i) {
    xr[i] = x[bidx * DN + lane + 32 * i];
    gu[i] = gum[bidx * DN + lane + 32 * i];
  }

  const float spb = wave_sum(softplusf(hb[s][lane]));  // sum_j softplus(h_j)

  // forward logits: Z_forward and Gumbel-argmax
  float zf = 0.f, bestv = -3.4e38f;
  int   bestk = 0;
#pragma unroll
  for (int i = 0; i < 8; ++i) {
    const int   k     = lane + 32 * i;
    const float delta = 1.f - 2.f * xr[i];
    float acc = 0.f;
#pragma unroll
    for (int j = 0; j < HN; ++j)
      acc += softplusf(hb[s][j] + delta * Wt[j * DN + k]);
    const float logit = 0.5f * (delta * bl[k] + acc - spb);
    zf += __expf(logit);
    const float g   = -__logf(-__logf(gu[i] + 1e-9f) + 1e-9f);
    const float key = logit + g;
    if (key > bestv) { bestv = key; bestk = k; }   // first-max within lane
  }
  zf = wave_sum(zf);
#pragma unroll
  for (int m = 16; m > 0; m >>= 1) {               // argmax, lowest-index ties
    const float ov = __shfl_xor(bestv, m, 32);
    const int   ok = __shfl_xor(bestk, m, 32);
    if (ov > bestv || (ov == bestv && ok < bestk)) { bestv = ov; bestk = ok; }
  }
  const int   ks  = bestk;
  const float dls = 1.f - 2.f * x[bidx * DN + ks];

  // rank-1 update of h for the flipped state
  const float h2j = hb[s][lane] + dls * Wt[lane * DN + ks];
  h2[s][lane] = h2j;
  const float spb2 = wave_sum(softplusf(h2j));
  __syncthreads();   // uniform across block; makes h2 lanes visible

  // reverse logits: Z_reverse
  float zr = 0.f;
#pragma unroll
  for (int i = 0; i < 8; ++i) {
    const int   k     = lane + 32 * i;
    const float xd    = (k == ks) ? 1.f - xr[i] : xr[i];
    const float delta = 1.f - 2.f * xd;
    float acc = 0.f;
#pragma unroll
    for (int j = 0; j < HN; ++j)
      acc += softplusf(h2[s][j] + delta * Wt[j * DN + k]);
    zr += __expf(0.5f * (delta * bl[k] + acc - spb2));
  }
  zr = wave_sum(zr);

  // accept / reject and write result
  const float a = (zf / zr > accu[bidx]) ? 1.f : 0.f;
#pragma unroll
  for (int i = 0; i < 8; ++i) {
    const int   k  = lane + 32 * i;
    const float xd = (k == ks) ? 1.f - xr[i] : xr[i];
    out[bidx * DN + k] = a * xd + (1.f - a) * xr[i];
  }
}

extern "C" void kernel_launch(void* const* d_in, const int* in_sizes, int n_in,
                              void* d_out, int out_size, void* d_ws, size_t ws_size,
                              hipStream_t stream) {
  const float* x    = (const float*)d_in[0];
  const float* W    = (const float*)d_in[1];
  const float* bvec = (const float*)d_in[2];
  const float* cvec = (const float*)d_in[3];
  const float* gum  = (const float*)d_in[4];
  const float* accu = (const float*)d_in[5];
  float* out = (float*)d_out;
  (void)in_sizes; (void)n_in; (void)out_size; (void)d_ws; (void)ws_size;

  dim3 grid(BN / 16), block(512);
  lbmcmc_kernel<<<grid, block, 0, stream>>>(x, W, bvec, cvec, gum, accu, out);
}